// Conv1d_NN_50654844289696
// MI455X (gfx1250) — compile-verified
//
#include <hip/hip_runtime.h>

typedef __attribute__((ext_vector_type(2))) float v2f;
typedef __attribute__((ext_vector_type(8))) float v8f;

#define B_DIM   32
#define C_IN    16
#define C_OUT   32
#define N_DIM   2048
#define K_NN    3
#define WAVES   8          // waves per block, each owns a 16-row strip

typedef unsigned int       u32;
typedef unsigned long long u64;

// Monotone map float -> u32 (preserves ordering, handles negatives).
static __device__ __forceinline__ u32 ford(float f) {
    u32 u = __float_as_uint(f);
    return u ^ ((u32)((int)u >> 31) | 0x80000000u);
}

// Branchless insert of `key` into ascending (k0 <= k1 <= k2).
// Lowers to v_min_u64/v_max_u64 on gfx1250.
static __device__ __forceinline__ void insert3(u64& k0, u64& k1, u64& k2, u64 key) {
    u64 a = key;
    u64 lo0 = (a < k0) ? a : k0;  u64 hi0 = (a < k0) ? k0 : a;
    k0 = lo0; a = hi0;
    u64 lo1 = (a < k1) ? a : k1;  u64 hi1 = (a < k1) ? k1 : a;
    k1 = lo1; a = hi1;
    k2 = (a < k2) ? a : k2;
}

// 256 threads = 8 waves per block; each wave fuses for its 16-row strip:
//   shared column-norm pre-pass -> gram (fp32 WMMA 16x16x4) -> rank -> top-3 -> gather -> conv.
__global__ __launch_bounds__(256)
void conv1d_nn_fused_kernel(const float* __restrict__ x,
                            const float* __restrict__ W,
                            const float* __restrict__ bias,
                            float* __restrict__ out)
{
    const int tid  = threadIdx.x;
    const int lane = tid & 31;           // wave32 lane
    const int wid  = tid >> 5;           // wave id within block
    const int l15  = lane & 15;
    const int h    = lane >> 4;          // half-wave select
    const int m0   = (blockIdx.x * WAVES + wid) * 16;   // this wave's query strip
    const int b    = blockIdx.y;

    const float* __restrict__ xb = x + (size_t)b * C_IN * N_DIM;

    __shared__ float sqn[N_DIM];                        // ||col||^2, shared by all 8 waves (8 KB)
    __shared__ u64   sK[WAVES][32][24];                 // per-wave top-3 keys (48 KB)
    __shared__ int   topIdx[WAVES][16][K_NN];
    __shared__ float nb[WAVES][16][K_NN][C_IN];         // gathered neighbors (24 KB)

    // ---- Shared pre-pass: squared norms of every column (2048/256 exact, no divergence).
    for (int n = tid; n < N_DIM; n += 256) {
        float s = 0.0f;
        #pragma unroll
        for (int c = 0; c < C_IN; ++c) {
            float v = xb[(size_t)c * N_DIM + n];
            s = fmaf(v, v, s);
        }
        sqn[n] = s;
    }
    __syncthreads();

    // ---- Load A strip (rows m0..m0+15, 16 channels) in WMMA A layout.
    // A (16x4, MxK): lane L -> M = L&15, VGPR v -> K = cbase + v + 2*(L>>4)
    v2f af[4];
    #pragma unroll
    for (int q = 0; q < 4; ++q) {
        const int c0 = 4 * q + 2 * h;
        af[q].x = xb[(size_t)c0 * N_DIM + m0 + l15];
        af[q].y = xb[(size_t)(c0 + 1) * N_DIM + m0 + l15];
    }

    // ---- Running top-3 keys per (row v, column-class l15).
    // Rank by (||col||^2 - 2*gram): per-row shift by ||row||^2 never changes order,
    // and the merge only compares candidates of the same row.
    u64 k0[8], k1[8], k2[8];
    #pragma unroll
    for (int v = 0; v < 8; ++v) {
        k0[v] = ~0ull; k1[v] = ~0ull; k2[v] = ~0ull;
    }

    #pragma unroll 2
    for (int n0 = 0; n0 < N_DIM; n0 += 16) {
        // B tile (4x16, KxN): lane L -> N = n0 + (L&15), VGPR v -> K = cbase + v + 2*(L>>4)
        v2f bf[4];
        #pragma unroll
        for (int q = 0; q < 4; ++q) {
            const int c0 = 4 * q + 2 * h;
            bf[q].x = xb[(size_t)c0 * N_DIM + n0 + l15];
            bf[q].y = xb[(size_t)(c0 + 1) * N_DIM + n0 + l15];
        }
        const float csq = sqn[n0 + l15];     // one LDS read, no cross-lane dep

        // 16x16 Gram tile, K=16 contraction via 4x fp32 WMMA.
        v8f acc = {};
        #pragma unroll
        for (int q = 0; q < 4; ++q)
            acc = __builtin_amdgcn_wmma_f32_16x16x4_f32(
                false, af[q], false, bf[q], (short)0, acc, false, false);

        const u32 j = (u32)(n0 + l15);
        #pragma unroll
        for (int v = 0; v < 8; ++v) {
            float d = fmaf(-2.0f, acc[v], csq);          // rank key (row-shifted distance)
            u64 key = ((u64)ford(d) << 32) | j;
            insert3(k0[v], k1[v], k2[v], key);
        }
    }

    // ---- Merge 16 column-classes per row -> per-row top-3 indices (wave-private region).
    #pragma unroll
    for (int v = 0; v < 8; ++v) {
        sK[wid][lane][v * 3 + 0] = k0[v];
        sK[wid][lane][v * 3 + 1] = k1[v];
        sK[wid][lane][v * 3 + 2] = k2[v];
    }
    __syncthreads();

    if (lane < 16) {
        const int m = lane;
        const int v = m & 7;
        const int lbase = (m < 8) ? 0 : 16;   // which half-wave held row m
        u64 b0 = ~0ull, b1 = ~0ull, b2 = ~0ull;
        for (int l = 0; l < 16; ++l) {
            #pragma unroll
            for (int k = 0; k < 3; ++k)
                insert3(b0, b1, b2, sK[wid][lbase + l][v * 3 + k]);
        }
        topIdx[wid][m][0] = (int)(u32)b0;
        topIdx[wid][m][1] = (int)(u32)b1;
        topIdx[wid][m][2] = (int)(u32)b2;
    }
    __syncthreads();

    // ---- Gather the 3 neighbor columns per row into LDS: nb[wid][m][k][c].
    for (int m = 0; m < 16; ++m) {
        for (int e = lane; e < K_NN * C_IN; e += 32) {
            int k = e >> 4;
            int c = e & 15;
            nb[wid][m][k][c] = xb[(size_t)c * N_DIM + topIdx[wid][m][k]];
        }
    }
    __syncthreads();

    // ---- Conv1d (kernel=K, stride=K): lane = output channel.
    const int o = lane;
    float w[C_IN * K_NN];
    #pragma unroll
    for (int e = 0; e < C_IN * K_NN; ++e)
        w[e] = W[o * (C_IN * K_NN) + e];          // W[o][c][k], e = c*3 + k
    const float bo = bias[o];

    float* __restrict__ ob = out + ((size_t)b * C_OUT + o) * N_DIM + m0;
    for (int m = 0; m < 16; ++m) {
        float acc = bo;
        #pragma unroll
        for (int c = 0; c < C_IN; ++c) {
            #pragma unroll
            for (int k = 0; k < K_NN; ++k)
                acc = fmaf(nb[wid][m][k][c], w[c * 3 + k], acc);
        }
        ob[m] = acc;
    }
}

extern "C" void kernel_launch(void* const* d_in, const int* in_sizes, int n_in,
                              void* d_out, int out_size, void* d_ws, size_t ws_size,
                              hipStream_t stream) {
    (void)in_sizes; (void)n_in; (void)out_size; (void)d_ws; (void)ws_size;
    const float* x    = (const float*)d_in[0];  // (32, 16, 2048)
    const float* W    = (const float*)d_in[1];  // (32, 16, 3)
    const float* bias = (const float*)d_in[2];  // (32,)
    float* out = (float*)d_out;                 // (32, 32, 2048)

    dim3 grid(N_DIM / (16 * WAVES), B_DIM);   // 16 x 32 blocks, 8 waves each = 4096 waves
    dim3 block(32 * WAVES);                   // 256 threads
    conv1d_nn_fused_kernel<<<grid, block, 0, stream>>>(x, W, bias, out);
}